// MicroConv_47493748359387
// MI455X (gfx1250) — compile-verified
//
#include <hip/hip_runtime.h>
#include <math.h>

// Problem constants (match reference)
#define D_IN   256
#define HEADS  8
#define ODIM   16
#define HID    (HEADS * ODIM)   // 128
#define NEG_SLOPE 0.2f
#define A_PAD  4                // LDS row stride = 260 floats: 16B-aligned, bank-stride 4

typedef __attribute__((ext_vector_type(2))) float v2f;
typedef __attribute__((ext_vector_type(8))) float v8f;

// ---------------------------------------------------------------------------
// init: zero output, m = -inf, denom = 0  (grid-stride)
// ---------------------------------------------------------------------------
__global__ void gat_init(float* __restrict__ out, float* __restrict__ m,
                         float* __restrict__ denom, int out_n, int nk) {
  int i = blockIdx.x * blockDim.x + threadIdx.x;
  int stride = gridDim.x * blockDim.x;
  for (int j = i; j < out_n; j += stride) out[j] = 0.0f;
  for (int j = i; j < nk; j += stride) { m[j] = -INFINITY; denom[j] = 0.0f; }
}

// ---------------------------------------------------------------------------
// h = feat @ W^T + b  via V_WMMA_F32_16X16X4_F32 (full fp32 precision).
// Block = 256 threads = 8 waves; computes a 16-row x 128-col tile of h.
// Wave w owns columns [16w, 16w+16) == head w. K loop: 256/4 = 64 WMMA ops.
//
// A tile (16x256) is shared by all 8 waves -> staged once in LDS
// (cooperative float4 loads, padded stride 260 floats for bank spread).
// W slice per wave (16x256) is read exactly once -> stays on global/L2 path.
//
// A (16x4 f32, 2 VGPR): lane l<16: row l,    K = k0,k0+1
//                       lane>=16 : row l-16, K = k0+2,k0+3
// B (4x16 f32, 2 VGPR): B(k,n) = W[col0+n][k]; same K-pair split per lane.
// C/D (16x16 f32, 8 VGPR): VGPR j -> M = j + 8*(lane>>4), N = lane&15.
// ---------------------------------------------------------------------------
__global__ __launch_bounds__(256) void gat_linear_wmma(
    const float* __restrict__ feat,   // (N, 256)
    const float* __restrict__ W,      // (128, 256) row-major
    const float* __restrict__ bias,   // (128)
    float* __restrict__ h,            // (N, 128)
    int N) {
  __shared__ float As[16][D_IN + A_PAD];

  const int row0  = blockIdx.x * 16;
  const int t     = threadIdx.x;
  const int lane  = t & 31;
  const int wave  = t >> 5;             // 0..7 == head
  const int col0  = wave * 16;
  const int l16   = lane & 15;
  const int lhalf = lane >> 4;          // 0 or 1
  const int khalf = lhalf * 2;          // K-pair offset within the 4-wide step

  // --- stage A tile: 16 rows x 256 cols; each thread moves 16 floats -------
  {
    int lrow = t >> 4;                  // 0..15
    int gr = row0 + lrow;
    if (gr >= N) gr = N - 1;            // safety clamp (N % 16 == 0 in practice)
    int c0 = (t & 15) * 16;             // 16-float segment, 64B aligned
    const float4* __restrict__ gp = (const float4*)(feat + (size_t)gr * D_IN + c0);
    float4* sp = (float4*)&As[lrow][c0];
#pragma unroll
    for (int q = 0; q < 4; ++q) sp[q] = gp[q];
  }
  __syncthreads();

  const float* __restrict__ ap = &As[l16][khalf];
  const float* __restrict__ bp = W + (size_t)(col0 + l16) * D_IN + khalf;

  v8f c = {};
#pragma unroll 8
  for (int k0 = 0; k0 < D_IN; k0 += 4) {
    v2f a, b;
    a.x = ap[k0];     a.y = ap[k0 + 1];
    b.x = bp[k0];     b.y = bp[k0 + 1];
    // (neg_a, A, neg_b, B, c_mod, C, reuse_a, reuse_b)
    c = __builtin_amdgcn_wmma_f32_16x16x4_f32(false, a, false, b,
                                              (short)0, c, false, false);
  }

  const int col = col0 + l16;
  const float bv = bias[col];
  const int rbase = row0 + lhalf * 8;
  float* __restrict__ hp = h + (size_t)rbase * HID + col;
  if (row0 + 16 <= N) {                 // full tile: unguarded store run
#pragma unroll
    for (int j = 0; j < 8; ++j) hp[(size_t)j * HID] = c[j] + bv;
  } else {                              // ragged tile (not hit for N=50000)
    for (int j = 0; j < 8; ++j)
      if (rbase + j < N) hp[(size_t)j * HID] = c[j] + bv;
  }
}

// ---------------------------------------------------------------------------
// el / er: per (node, head) dot(h[node, head*16 : +16], attn[head, off : off+16])
// ---------------------------------------------------------------------------
__global__ void gat_logits(const float* __restrict__ h,
                           const float* __restrict__ attn,   // (8, 32)
                           int attn_off,                     // 0 or 16
                           float* __restrict__ out,          // (N, 8)
                           int N) {
  int idx = blockIdx.x * blockDim.x + threadIdx.x;
  if (idx >= N * HEADS) return;
  int node = idx >> 3;
  int head = idx & 7;
  const float* hp = h + (size_t)node * HID + head * ODIM;
  const float* ap = attn + head * (2 * ODIM) + attn_off;
  float s = 0.0f;
#pragma unroll
  for (int d = 0; d < ODIM; ++d) s += hp[d] * ap[d];
  out[idx] = s;
}

// ---------------------------------------------------------------------------
// Edge score: e = LeakyReLU(el[src] + er[dst]); running segment-max via
// global_atomic_max_num_f32.
// ---------------------------------------------------------------------------
__global__ void gat_edge_score(const int* __restrict__ src,
                               const int* __restrict__ dst,
                               const float* __restrict__ el,
                               const float* __restrict__ er,
                               float* __restrict__ ebuf,     // (E, 8)
                               float* __restrict__ m,        // (Nd, 8)
                               int E) {
  int idx = blockIdx.x * blockDim.x + threadIdx.x;
  if (idx >= E * HEADS) return;
  int e = idx >> 3;
  int h = idx & 7;
  int s = src[e], d = dst[e];
  float v = el[s * HEADS + h] + er[d * HEADS + h];
  v = (v >= 0.0f) ? v : NEG_SLOPE * v;
  ebuf[idx] = v;
  atomicMax(&m[d * HEADS + h], v);
}

// ---------------------------------------------------------------------------
// e_exp = exp(e - m[dst]); denom[dst] += e_exp (global_atomic_add_f32)
// ---------------------------------------------------------------------------
__global__ void gat_edge_exp(const int* __restrict__ dst,
                             const float* __restrict__ m,
                             float* __restrict__ ebuf,
                             float* __restrict__ denom,
                             int E) {
  int idx = blockIdx.x * blockDim.x + threadIdx.x;
  if (idx >= E * HEADS) return;
  int e = idx >> 3;
  int h = idx & 7;
  int d = dst[e];
  float w = __expf(ebuf[idx] - m[d * HEADS + h]);
  ebuf[idx] = w;
  atomicAdd(&denom[d * HEADS + h], w);
}

// ---------------------------------------------------------------------------
// out[dst, head, :] += (e_exp / denom[dst, head]) * hs[src, head, :]
// ---------------------------------------------------------------------------
__global__ void gat_aggregate(const int* __restrict__ src,
                              const int* __restrict__ dst,
                              const float* __restrict__ ebuf,
                              const float* __restrict__ denom,
                              const float* __restrict__ hs,
                              float* __restrict__ out,
                              int E) {
  int idx = blockIdx.x * blockDim.x + threadIdx.x;
  if (idx >= E * HEADS) return;
  int e = idx >> 3;
  int h = idx & 7;
  int s = src[e], d = dst[e];
  float a = ebuf[idx] / denom[d * HEADS + h];
  const float* hp = hs + (size_t)s * HID + h * ODIM;
  float* op = out + (size_t)d * HID + h * ODIM;
#pragma unroll
  for (int t = 0; t < ODIM; ++t) atomicAdd(&op[t], a * hp[t]);
}

// ---------------------------------------------------------------------------
// kernel_launch
// inputs: 0 feat_src, 1 feat_dst, 2 W_src, 3 b_src, 4 W_dst, 5 b_dst,
//         6 attn, 7 src_idx, 8 dst_idx
// ---------------------------------------------------------------------------
extern "C" void kernel_launch(void* const* d_in, const int* in_sizes, int n_in,
                              void* d_out, int out_size, void* d_ws, size_t ws_size,
                              hipStream_t stream) {
  const float* feat_src = (const float*)d_in[0];
  const float* feat_dst = (const float*)d_in[1];
  const float* W_src    = (const float*)d_in[2];
  const float* b_src    = (const float*)d_in[3];
  const float* W_dst    = (const float*)d_in[4];
  const float* b_dst    = (const float*)d_in[5];
  const float* attn     = (const float*)d_in[6];
  const int*   src_idx  = (const int*)d_in[7];
  const int*   dst_idx  = (const int*)d_in[8];
  float* out = (float*)d_out;

  const int Ns = in_sizes[0] / D_IN;       // 50000
  const int Nd = in_sizes[1] / D_IN;       // 50000
  const int E  = in_sizes[7];              // 800000

  // Workspace layout (fp32)
  float* ws = (float*)d_ws;
  float* hs    = ws;                                   // Ns*128
  float* hd    = hs + (size_t)Ns * HID;                // Nd*128
  float* el    = hd + (size_t)Nd * HID;                // Ns*8
  float* er    = el + (size_t)Ns * HEADS;              // Nd*8
  float* m     = er + (size_t)Nd * HEADS;              // Nd*8
  float* denom = m  + (size_t)Nd * HEADS;              // Nd*8
  float* ebuf  = denom + (size_t)Nd * HEADS;           // E*8

  // 1. init (out=0, m=-inf, denom=0)
  gat_init<<<1024, 256, 0, stream>>>(out, m, denom, Nd * HID, Nd * HEADS);

  // 2. node linear transforms via fp32 WMMA (A tile staged through LDS)
  gat_linear_wmma<<<(Ns + 15) / 16, 256, 0, stream>>>(feat_src, W_src, b_src, hs, Ns);
  gat_linear_wmma<<<(Nd + 15) / 16, 256, 0, stream>>>(feat_dst, W_dst, b_dst, hd, Nd);

  // 3. attention logits
  {
    int n = Ns * HEADS;
    gat_logits<<<(n + 255) / 256, 256, 0, stream>>>(hs, attn, 0, el, Ns);
    n = Nd * HEADS;
    gat_logits<<<(n + 255) / 256, 256, 0, stream>>>(hd, attn, ODIM, er, Nd);
  }

  // 4. edge softmax + aggregation
  {
    int n = E * HEADS;
    int g = (n + 255) / 256;
    gat_edge_score<<<g, 256, 0, stream>>>(src_idx, dst_idx, el, er, ebuf, m, E);
    gat_edge_exp<<<g, 256, 0, stream>>>(dst_idx, m, ebuf, denom, E);
    gat_aggregate<<<g, 256, 0, stream>>>(src_idx, dst_idx, ebuf, denom, hs, out, E);
  }
}